// RegionalAttentionPool_80358838108790
// MI455X (gfx1250) — compile-verified
//
#include <hip/hip_runtime.h>

// Problem constants (from the reference): B=16, N=4096, D=512, R=64, K=128
#define BB 16
#define NN 4096
#define DD 512
#define RR 64
#define KK 128

typedef __attribute__((ext_vector_type(2))) float    v2f;
typedef __attribute__((ext_vector_type(8))) float    v8f;
typedef __attribute__((ext_vector_type(4))) unsigned v4u;
typedef __attribute__((ext_vector_type(8))) unsigned v8u;

// One workgroup handles one (b, r) pair:
//   1) TDM gather-mode: 8 descriptors x 16 indexed rows -> 256 KB LDS tile
//   2) scores = tile . W + bias   (8 waves x 16 rows, shuffle reduction)
//   3) softmax over K=128         (wave 0)
//   4) out = attn . tile          (V_WMMA_F32_16X16X4_F32, f32 accumulate)
__global__ __launch_bounds__(256, 1)
void RegionalAttentionPool_kernel(const float* __restrict__ x,
                                  const int*   __restrict__ ridx,
                                  const float* __restrict__ W,
                                  const float* __restrict__ bias,
                                  float*       __restrict__ out) {
  __shared__ float tile[KK][DD];   // 256 KB gathered rows (fits 320 KB/WGP LDS)
  __shared__ float wlds[DD];       // 2 KB: W staged for reuse
  __shared__ float sc[KK];         // scores -> attention weights

  const int t    = threadIdx.x;
  const int lane = t & 31;
  const int wave = t >> 5;
  const int bb   = blockIdx.x / RR;
  const int r    = blockIdx.x % RR;

  // Stage W into LDS (all threads; overlaps with wave 0's descriptor setup).
  for (int i = t; i < DD; i += 256) wlds[i] = W[i];

  // ---- Phase 1: TDM gather (wave 0 issues 8 gather-mode descriptors) ----
  if (wave == 0) {
    const unsigned lds_base = (unsigned)(uintptr_t)(&tile[0][0]); // LDS byte offset
    const unsigned long long gaddr =
        (unsigned long long)(uintptr_t)(x + (size_t)bb * NN * DD); // x[bb] base
    const int* ib = ridx + r * KK;

    for (int c = 0; c < 8; ++c) {
      // D# group 0: count=1, gather_mode=1, 16-bit indices, lds_addr, global_addr, type=2
      v4u g0;
      g0[0] = 0x80000001u;                                  // count=1 | gather_mode(31)
      g0[1] = lds_base + (unsigned)c * (16u * DD * 4u);     // 32 KB per descriptor
      g0[2] = (unsigned)gaddr;                              // global_addr[31:0]
      g0[3] = ((unsigned)(gaddr >> 32) & 0x01FFFFFFu) | (2u << 30); // addr[56:32] | type=2
      // D# group 1: data_size=4B, tensor_dim0=512, tensor_dim1=4096,
      //             tile_dim0=512, tile_dim1=16 indices, dim0_stride=512
      v8u g1;
      g1[0] = 0x00020000u;                 // workgroup_mask=0, data_size=2 (4 bytes)
      g1[1] = (unsigned)DD << 16;          // tensor_dim0[15:0] at bits 63:48
      g1[2] = (unsigned)NN << 16;          // tensor_dim1[15:0] at bits 95:80
      g1[3] = (unsigned)DD << 16;          // tile_dim0 at bits 127:112
      g1[4] = 16u;                         // tile_dim1 = #valid row indices
      g1[5] = (unsigned)DD;                // tensor_dim0_stride[31:0]
      g1[6] = 0u;
      g1[7] = 0u;
      // D# groups 2/3: sixteen 16-bit row indices
      const int* p = ib + c * 16;
      v4u g2, g3;
      #pragma unroll
      for (int j = 0; j < 4; ++j)
        g2[j] = ((unsigned)p[2 * j] & 0xFFFFu) | (((unsigned)p[2 * j + 1] & 0xFFFFu) << 16);
      #pragma unroll
      for (int j = 0; j < 4; ++j)
        g3[j] = ((unsigned)p[8 + 2 * j] & 0xFFFFu) | (((unsigned)p[8 + 2 * j + 1] & 0xFFFFu) << 16);

      asm volatile("tensor_load_to_lds %0, %1, %2, %3"
                   :: "s"(g0), "s"(g1), "s"(g2), "s"(g3)
                   : "memory");
    }
    __builtin_amdgcn_s_wait_tensorcnt(0);  // TENSORcnt == 0 -> tile resident in LDS
  }
  __syncthreads();

  // ---- Phase 2: scores[k] = tile[k] . W + bias (wave w owns rows 16w..16w+15) ----
  const float bval = bias[0];
  for (int k = wave * 16; k < wave * 16 + 16; ++k) {
    float acc = 0.f;
    #pragma unroll
    for (int j = 0; j < DD / 32; ++j) {     // lane-contiguous: conflict-free LDS reads
      const int d = lane + 32 * j;
      acc += tile[k][d] * wlds[d];
    }
    #pragma unroll
    for (int off = 16; off > 0; off >>= 1)  // 32-lane (wave32) reduction
      acc += __shfl_xor(acc, off, 32);
    if (lane == 0) sc[k] = acc + bval;
  }
  __syncthreads();

  // ---- Phase 3: softmax over K=128 (wave 0; 4 values per lane) ----
  if (wave == 0) {
    float v[4];
    float m = -3.402823466e38f;
    #pragma unroll
    for (int j = 0; j < 4; ++j) { v[j] = sc[lane + 32 * j]; m = fmaxf(m, v[j]); }
    #pragma unroll
    for (int off = 16; off > 0; off >>= 1) m = fmaxf(m, __shfl_xor(m, off, 32));
    float s = 0.f;
    #pragma unroll
    for (int j = 0; j < 4; ++j) { v[j] = __expf(v[j] - m); s += v[j]; }
    #pragma unroll
    for (int off = 16; off > 0; off >>= 1) s += __shfl_xor(s, off, 32);
    const float inv = 1.f / s;
    #pragma unroll
    for (int j = 0; j < 4; ++j) sc[lane + 32 * j] = v[j] * inv;
  }
  __syncthreads();

  // ---- Phase 4: out = attn . tile via V_WMMA_F32_16X16X4_F32 ----
  // A (16x4 f32): attn chunk replicated down the 16 M-rows.
  //   VGPR0 = A[M=lane&15][K = lane<16 ? 0 : 2], VGPR1 = K=1 / K=3.
  // B (4x16 f32): Xr chunk, B[kk][j] = tile[k0+kk][d0+j].
  //   VGPR0 = B[K = lane<16 ? 0 : 2][N=lane&15], VGPR1 = next K row.
  // D row 0 (VGPR0, lanes 0-15) = 16 accumulated outputs.
  const int half = lane >> 4;     // 0 -> lanes 0-15, 1 -> lanes 16-31
  const int col  = lane & 15;
  float* orow = out + (size_t)blockIdx.x * DD;

  for (int chunk = 0; chunk < 4; ++chunk) {
    const int d0 = (wave * 4 + chunk) * 16;
    v8f acc = {0.f, 0.f, 0.f, 0.f, 0.f, 0.f, 0.f, 0.f};
    #pragma unroll 8
    for (int k0 = 0; k0 < KK; k0 += 4) {
      v2f a, bm;
      a[0]  = sc[k0 + 2 * half + 0];
      a[1]  = sc[k0 + 2 * half + 1];
      bm[0] = tile[k0 + 2 * half + 0][d0 + col];
      bm[1] = tile[k0 + 2 * half + 1][d0 + col];
      acc = __builtin_amdgcn_wmma_f32_16x16x4_f32(
          /*neg_a=*/false, a, /*neg_b=*/false, bm,
          /*c_mod=*/(short)0, acc, /*reuse_a=*/false, /*reuse_b=*/false);
    }
    if (half == 0) orow[d0 + col] = acc[0];
  }
}

extern "C" void kernel_launch(void* const* d_in, const int* in_sizes, int n_in,
                              void* d_out, int out_size, void* d_ws, size_t ws_size,
                              hipStream_t stream) {
  const float* x    = (const float*)d_in[0];   // (B, N, D) f32
  const int*   ridx = (const int*)d_in[1];     // (R, K) i32
  const float* W    = (const float*)d_in[2];   // (1, D) f32
  const float* bias = (const float*)d_in[3];   // (1,) f32
  float*       out  = (float*)d_out;           // (B, R, D) f32

  dim3 grid(BB * RR);   // one workgroup per (b, r)
  dim3 block(256);      // 8 wave32 waves
  RegionalAttentionPool_kernel<<<grid, block, 0, stream>>>(x, ridx, W, bias, out);
}